// Retinanet_40699110097482
// MI455X (gfx1250) — compile-verified
//
#include <hip/hip_runtime.h>

#define NB        2
#define NCLS      80
#define K_PRE     500
#define K_PAD     512
#define CAND_CAP  1024
#define MAX_DET   300
#define SCORE_THR 0.05f
#define NMS_THR   0.5f
#define MIN_SIZE  0.01f
#define BBOX_CLAMP 4.135166556742356f  /* log(1000/16) */

typedef __attribute__((ext_vector_type(2))) float v2f;
typedef __attribute__((ext_vector_type(8))) float v8f;

// ---------------------------------------------------------------------------
// Kernel 1: decode + clip + min-size validity.  One thread per (b,a).
// ---------------------------------------------------------------------------
__global__ void rn_decode(const float* __restrict__ deltas,
                          const float* __restrict__ anchors,
                          const int* __restrict__ p_h,
                          const int* __restrict__ p_w,
                          float* __restrict__ boxes,
                          unsigned* __restrict__ valid,
                          int n) {
  int i = blockIdx.x * blockDim.x + threadIdx.x;
  if (i >= n) return;
  float W = (float)(*p_w);
  float H = (float)(*p_h);
  float a0 = anchors[(size_t)i * 4 + 0];
  float a1 = anchors[(size_t)i * 4 + 1];
  float a2 = anchors[(size_t)i * 4 + 2];
  float a3 = anchors[(size_t)i * 4 + 3];
  float aw = a2 - a0, ah = a3 - a1;
  float acx = a0 + 0.5f * aw, acy = a1 + 0.5f * ah;
  float dx = deltas[(size_t)i * 4 + 0];
  float dy = deltas[(size_t)i * 4 + 1];
  float dw = fminf(deltas[(size_t)i * 4 + 2], BBOX_CLAMP);
  float dh = fminf(deltas[(size_t)i * 4 + 3], BBOX_CLAMP);
  float pcx = dx * aw + acx;
  float pcy = dy * ah + acy;
  float pw  = expf(dw) * aw;
  float ph  = expf(dh) * ah;
  float x1 = fminf(fmaxf(pcx - 0.5f * pw, 0.0f), W);
  float y1 = fminf(fmaxf(pcy - 0.5f * ph, 0.0f), H);
  float x2 = fminf(fmaxf(pcx + 0.5f * pw, 0.0f), W);
  float y2 = fminf(fmaxf(pcy + 0.5f * ph, 0.0f), H);
  boxes[(size_t)i * 4 + 0] = x1;
  boxes[(size_t)i * 4 + 1] = y1;
  boxes[(size_t)i * 4 + 2] = x2;
  boxes[(size_t)i * 4 + 3] = y2;
  valid[i] = (((x2 - x1) >= MIN_SIZE) && ((y2 - y1) >= MIN_SIZE)) ? 1u : 0u;
}

// ---------------------------------------------------------------------------
// Kernel 2: per-(b,c) top-K via 1024-bin histogram threshold + bitonic sort.
// One 256-thread block per class-image.
// ---------------------------------------------------------------------------
__global__ void rn_topk(const float* __restrict__ scores,
                        const float* __restrict__ boxes,
                        const unsigned* __restrict__ valid,
                        float* __restrict__ cscore,
                        float* __restrict__ cbox,
                        int A) {
  __shared__ unsigned s_hist[1024];
  __shared__ float    s_cs[CAND_CAP];
  __shared__ int      s_ci[CAND_CAP];
  __shared__ int      s_T;
  __shared__ unsigned s_cnt;

  const int bc  = blockIdx.x;            // 0..159
  const int b   = bc / NCLS;
  const int c   = bc % NCLS;
  const int tid = threadIdx.x;

  for (int t = tid; t < 1024; t += 256) s_hist[t] = 0u;
  __syncthreads();

  const float*    sbase = scores + ((size_t)b * A) * NCLS + c;  // stride NCLS
  const unsigned* vbase = valid + (size_t)b * A;

  // Pass 1: histogram of valid scores above threshold.
  for (int a = tid; a < A; a += 256) {
    __builtin_prefetch(sbase + (size_t)(a + 1024) * NCLS, 0, 0);
    float s = sbase[(size_t)a * NCLS];
    if (vbase[a] && s > SCORE_THR) {
      int bin = (int)(s * 1024.0f);
      bin = bin < 0 ? 0 : (bin > 1023 ? 1023 : bin);
      atomicAdd(&s_hist[bin], 1u);
    }
  }
  __syncthreads();

  // Find threshold bin: largest bin T with suffix-count >= K_PRE (0 if none).
  if (tid == 0) {
    unsigned cum = 0;
    int T = 0;
    for (int bin = 1023; bin >= 0; --bin) {
      cum += s_hist[bin];
      if (cum >= (unsigned)K_PRE) { T = bin; break; }
    }
    s_T = T;
    s_cnt = 0u;
  }
  __syncthreads();
  const int T = s_T;

  // Pass 2: collect candidates at/above threshold bin.
  for (int a = tid; a < A; a += 256) {
    float s = sbase[(size_t)a * NCLS];
    if (vbase[a] && s > SCORE_THR) {
      int bin = (int)(s * 1024.0f);
      bin = bin < 0 ? 0 : (bin > 1023 ? 1023 : bin);
      if (bin >= T) {
        unsigned p = atomicAdd(&s_cnt, 1u);
        if (p < CAND_CAP) { s_cs[p] = s; s_ci[p] = a; }
      }
    }
  }
  __syncthreads();

  unsigned cnt = s_cnt; if (cnt > CAND_CAP) cnt = CAND_CAP;
  for (int p = tid; p < CAND_CAP; p += 256)
    if ((unsigned)p >= cnt) { s_cs[p] = -1.0f; s_ci[p] = 0; }
  __syncthreads();

  // Bitonic sort, descending by score.
  for (unsigned k = 2; k <= CAND_CAP; k <<= 1) {
    for (unsigned j = k >> 1; j > 0; j >>= 1) {
      for (unsigned t = tid; t < CAND_CAP; t += 256) {
        unsigned ixj = t ^ j;
        if (ixj > t) {
          bool desc = ((t & k) == 0);
          float s0 = s_cs[t], s1 = s_cs[ixj];
          bool swp = desc ? (s0 < s1) : (s0 > s1);
          if (swp) {
            int i0 = s_ci[t];
            s_cs[t] = s1; s_cs[ixj] = s0;
            s_ci[t] = s_ci[ixj]; s_ci[ixj] = i0;
          }
        }
      }
      __syncthreads();
    }
  }

  // Emit K_PAD candidates (entries >= K_PRE forced invalid, matching ref).
  for (int p = tid; p < K_PAD; p += 256) {
    float s = (p < K_PRE) ? s_cs[p] : -1.0f;
    int   a = s_ci[p];
    size_t o = ((size_t)bc * K_PAD + p);
    cscore[o] = s;
    const float* bp = boxes + ((size_t)b * A + a) * 4;
    cbox[o * 4 + 0] = bp[0];
    cbox[o * 4 + 1] = bp[1];
    cbox[o * 4 + 2] = bp[2];
    cbox[o * 4 + 3] = bp[3];
  }
}

// ---------------------------------------------------------------------------
// Kernel 3: per-(b,c) NMS.  256 threads = 8 wave32s.  Suppression bitmask
// built tile-wise: V_WMMA_F32_16X16X4_F32 produces the pairwise union base
// (area_i + area_j) for a 16x16 tile; VALU computes intersection.  The
// IoU>thr predicate is evaluated division-free:  inter/uni > t  <=>
// inter > t*uni  (uni clamped positive), eliminating the IEEE div sequence
// from the inner loop.  ds_or accumulates the suppression bitmask; greedy
// keep-propagation runs over the LDS bitmask in score order.
// ---------------------------------------------------------------------------
__global__ void rn_nms(const float* __restrict__ cscore,
                       const float* __restrict__ cbox,
                       float* __restrict__ out) {
  __shared__ float    s_bx[K_PAD * 4];      //  8 KB
  __shared__ float    s_ar[K_PAD];          //  2 KB
  __shared__ float    s_sc[K_PAD];          //  2 KB
  __shared__ unsigned s_sup[K_PAD * 16];    // 32 KB: row i -> 512-bit mask
  __shared__ unsigned s_keep[16];           // 512-bit keep mask
  __shared__ int      s_pos[K_PAD];
  __shared__ int      s_nk;

  const int bc  = blockIdx.x;
  const int tid = threadIdx.x;
  const int lane = tid & 31;
  const int wave = tid >> 5;

  // Load candidates.
  for (int p = tid; p < K_PAD; p += 256) {
    size_t o = (size_t)bc * K_PAD + p;
    float x1 = cbox[o * 4 + 0], y1 = cbox[o * 4 + 1];
    float x2 = cbox[o * 4 + 2], y2 = cbox[o * 4 + 3];
    s_bx[p * 4 + 0] = x1; s_bx[p * 4 + 1] = y1;
    s_bx[p * 4 + 2] = x2; s_bx[p * 4 + 3] = y2;
    s_ar[p] = (x2 - x1) * (y2 - y1);
    s_sc[p] = cscore[o];
  }
  for (int w = tid; w < K_PAD * 16; w += 256) s_sup[w] = 0u;
  if (tid < 16) s_keep[tid] = 0u;
  __syncthreads();
  for (int p = tid; p < K_PAD; p += 256)
    if (s_sc[p] > SCORE_THR) atomicOr(&s_keep[p >> 5], 1u << (p & 31));
  __syncthreads();

  // 32x32 grid of 16x16 tiles; one tile per wave iteration.
  for (int tile = wave; tile < 1024; tile += 8) {
    const int i0 = (tile >> 5) * 16;   // row block
    const int j0 = (tile & 31) * 16;   // col block

    // A (16x4): row m = [area_i(m), 1, 0, 0]; B (4x16): col n = [1, area_j(n), 0, 0]^T
    v2f afrag, bfrag;
    afrag.x = (lane < 16) ? s_ar[i0 + lane] : 0.0f;
    afrag.y = (lane < 16) ? 1.0f : 0.0f;
    bfrag.x = (lane < 16) ? 1.0f : 0.0f;
    bfrag.y = (lane < 16) ? s_ar[j0 + (lane & 15)] : 0.0f;
    v8f acc = {};
    acc = __builtin_amdgcn_wmma_f32_16x16x4_f32(
        /*neg_a=*/false, afrag, /*neg_b=*/false, bfrag,
        /*c_mod=*/(short)0, acc, /*reuse_a=*/false, /*reuse_b=*/false);

    const int n = j0 + (lane & 15);
    const float nx1 = s_bx[n * 4 + 0], ny1 = s_bx[n * 4 + 1];
    const float nx2 = s_bx[n * 4 + 2], ny2 = s_bx[n * 4 + 3];
    #pragma unroll
    for (int v = 0; v < 8; ++v) {
      const int m = i0 + v + ((lane >= 16) ? 8 : 0);
      float ix1 = fmaxf(s_bx[m * 4 + 0], nx1);
      float iy1 = fmaxf(s_bx[m * 4 + 1], ny1);
      float ix2 = fminf(s_bx[m * 4 + 2], nx2);
      float iy2 = fminf(s_bx[m * 4 + 3], ny2);
      float iw = fmaxf(ix2 - ix1, 0.0f);
      float ih = fmaxf(iy2 - iy1, 0.0f);
      float inter = iw * ih;
      float uni = fmaxf(acc[v] - inter, 1e-9f);  // area_m + area_n from WMMA
      // inter/uni > NMS_THR  <=>  inter > NMS_THR * uni  (uni > 0)
      if (n > m && inter > NMS_THR * uni)
        atomicOr(&s_sup[m * 16 + (n >> 5)], 1u << (n & 31));
    }
  }
  __syncthreads();

  // Greedy suppression in score order (bits only exist for j > i).
  for (int i = 0; i < K_PAD; ++i) {
    bool alive = (s_keep[i >> 5] >> (i & 31)) & 1u;
    __syncthreads();
    if (alive && tid < 16) s_keep[tid] &= ~s_sup[i * 16 + tid];
    __syncthreads();
  }

  // Compact: kept detections first, score order preserved.
  if (tid == 0) {
    int k = 0;
    for (int p = 0; p < K_PAD; ++p) {
      bool kv = (s_keep[p >> 5] >> (p & 31)) & 1u;
      s_pos[p] = kv ? k++ : -1;
    }
    s_nk = k;
  }
  __syncthreads();

  float* o = out + (size_t)bc * MAX_DET * 5;
  for (int p = tid; p < K_PAD; p += 256) {
    int pos = s_pos[p];
    if (pos >= 0 && pos < MAX_DET) {
      o[pos * 5 + 0] = s_bx[p * 4 + 0];
      o[pos * 5 + 1] = s_bx[p * 4 + 1];
      o[pos * 5 + 2] = s_bx[p * 4 + 2];
      o[pos * 5 + 3] = s_bx[p * 4 + 3];
      o[pos * 5 + 4] = s_sc[p];
    }
  }
  int nk = s_nk; if (nk > MAX_DET) nk = MAX_DET;
  for (int r = tid; r < MAX_DET; r += 256) {
    if (r >= nk) {
      o[r * 5 + 0] = 0.0f; o[r * 5 + 1] = 0.0f;
      o[r * 5 + 2] = 0.0f; o[r * 5 + 3] = 0.0f;
      o[r * 5 + 4] = -1.0f;
    }
  }
}

// ---------------------------------------------------------------------------
extern "C" void kernel_launch(void* const* d_in, const int* in_sizes, int n_in,
                              void* d_out, int out_size, void* d_ws, size_t ws_size,
                              hipStream_t stream) {
  (void)n_in; (void)out_size; (void)ws_size;
  const float* deltas  = (const float*)d_in[0];
  const float* scores  = (const float*)d_in[1];
  const float* anchors = (const float*)d_in[2];
  const int*   p_h     = (const int*)d_in[3];
  const int*   p_w     = (const int*)d_in[4];

  const int A   = in_sizes[0] / (NB * 4);
  const int nBA = NB * A;

  char* ws = (char*)d_ws;
  size_t off = 0;
  float*    boxes  = (float*)(ws + off);    off += (size_t)nBA * 4 * sizeof(float);
  unsigned* valid  = (unsigned*)(ws + off); off += (size_t)nBA * sizeof(unsigned);
  float*    cscore = (float*)(ws + off);    off += (size_t)NB * NCLS * K_PAD * sizeof(float);
  float*    cbox   = (float*)(ws + off);

  rn_decode<<<(nBA + 255) / 256, 256, 0, stream>>>(deltas, anchors, p_h, p_w,
                                                   boxes, valid, nBA);
  rn_topk<<<NB * NCLS, 256, 0, stream>>>(scores, boxes, valid, cscore, cbox, A);
  rn_nms<<<NB * NCLS, 256, 0, stream>>>(cscore, cbox, (float*)d_out);
}